// GCNConv_27771258536567
// MI455X (gfx1250) — compile-verified
//
#include <hip/hip_runtime.h>

// ---------------------------------------------------------------------------
// GCNConv on MI455X (gfx1250):
//   Phase 1: H = X @ W^T  via V_WMMA_F32_16X16X4_F32 (fp32 tensor path).
//            Each wave32 owns a 16-row stripe and computes ALL four 16x16
//            N-tiles (4 accumulators), so X is read from HBM exactly once.
//            W (16 KB) is staged in LDS and feeds B fragments (ds_load_b64).
//   Phase 2: out[rows[e]] += edge_weight[e] * H[cols[e]]  (wave-per-edge,
//            f32 global atomics; H = 12.8 MB so gathers are L2-resident
//            against the 192 MB L2).
// ---------------------------------------------------------------------------

typedef float v2f __attribute__((ext_vector_type(2)));
typedef float v8f __attribute__((ext_vector_type(8)));

#define IN_DIM  64
#define OUT_DIM 64
#define NTILES  (OUT_DIM / 16)   // 4

// -------------------------- zero the output --------------------------------
__global__ void zero_out_kernel(float4* __restrict__ out, int n4) {
    int i = blockIdx.x * blockDim.x + threadIdx.x;
    if (i < n4) out[i] = make_float4(0.f, 0.f, 0.f, 0.f);
}

// -------------------- Phase 1: H = X * W^T  (WMMA f32) ---------------------
// Block = 128 threads = 4 wave32s; wave w handles rows (blockIdx.x*4+w)*16.
// Per K-step (K=4): one A fragment load feeds 4 WMMAs (one per N tile).
//
// A-fragment (32-bit 16x4): lane L<16: row=L,    K={k0,k0+1}
//                           lane L>=16: row=L-16, K={k0+2,k0+3}
// B mirrors with N across lanes: B[k][n] = W[n][k]   (h = x @ W^T)
// C/D (32-bit 16x16): VGPR v -> row = v + (L>>4)*8, col = L & 15.
__global__ void __launch_bounds__(128)
gcn_gemm_wmma_kernel(const float* __restrict__ X,    // [n_nodes, 64]
                     const float* __restrict__ W,    // [64, 64]  (out, in)
                     float* __restrict__ H,          // [n_nodes, 64]
                     int n_nodes) {
    __shared__ float sW[OUT_DIM * IN_DIM];           // 16 KB of 320 KB WGP LDS

    // Cooperative stage of W into LDS (4096 floats / 128 thr, float4).
    {
        const float4* src = reinterpret_cast<const float4*>(W);
        float4*       dst = reinterpret_cast<float4*>(sW);
        for (int i = threadIdx.x; i < (OUT_DIM * IN_DIM) / 4; i += 128)
            dst[i] = src[i];
    }
    __syncthreads();

    const int lane  = threadIdx.x & 31;
    const int wave  = threadIdx.x >> 5;
    const int m     = lane & 15;
    const int khalf = (lane >> 4) * 2;               // {0, 2}

    const int rowBase = (blockIdx.x * 4 + wave) * 16;

    // Branch-free clamp: EXEC must stay all-1s across the WMMA region.
    int row = rowBase + m;
    if (row >= n_nodes) row = n_nodes - 1;

    const float* arow = X + (size_t)row * IN_DIM;

    v8f acc0 = {}, acc1 = {}, acc2 = {}, acc3 = {};

#pragma unroll
    for (int k0 = 0; k0 < IN_DIM; k0 += 4) {
        v2f a = *reinterpret_cast<const v2f*>(arow + k0 + khalf);       // global_load_b64
        v2f b0 = *reinterpret_cast<const v2f*>(sW + ( 0 + m) * IN_DIM + k0 + khalf); // ds_load_b64
        v2f b1 = *reinterpret_cast<const v2f*>(sW + (16 + m) * IN_DIM + k0 + khalf);
        v2f b2 = *reinterpret_cast<const v2f*>(sW + (32 + m) * IN_DIM + k0 + khalf);
        v2f b3 = *reinterpret_cast<const v2f*>(sW + (48 + m) * IN_DIM + k0 + khalf);
        acc0 = __builtin_amdgcn_wmma_f32_16x16x4_f32(false, a, false, b0, (short)0, acc0, false, false);
        acc1 = __builtin_amdgcn_wmma_f32_16x16x4_f32(false, a, false, b1, (short)0, acc1, false, false);
        acc2 = __builtin_amdgcn_wmma_f32_16x16x4_f32(false, a, false, b2, (short)0, acc2, false, false);
        acc3 = __builtin_amdgcn_wmma_f32_16x16x4_f32(false, a, false, b3, (short)0, acc3, false, false);
    }

    // ---- store the 16x64 result stripe ----
    const int mhi  = (lane >> 4) * 8;
    const int ncol = lane & 15;
    float* hbase = H + (size_t)(rowBase + mhi) * OUT_DIM + ncol;

    if (rowBase + 16 <= n_nodes) {
        // Full tile: straight-line stores, single wave-uniform branch.
#pragma unroll
        for (int v = 0; v < 8; ++v) {
            float* p = hbase + (size_t)v * OUT_DIM;
            p[ 0] = acc0[v];
            p[16] = acc1[v];
            p[32] = acc2[v];
            p[48] = acc3[v];
        }
    } else {
#pragma unroll
        for (int v = 0; v < 8; ++v) {
            int r = rowBase + mhi + v;
            if (r < n_nodes) {
                float* p = H + (size_t)r * OUT_DIM + ncol;
                p[ 0] = acc0[v];
                p[16] = acc1[v];
                p[32] = acc2[v];
                p[48] = acc3[v];
            }
        }
    }
}

// ------------- Phase 2: scatter-add, one wave32 per edge -------------------
// Each lane handles 2 of the 64 output dims -> each wave's atomics form one
// contiguous 256 B burst on the destination row (friendly to L2 atomic units).
__global__ void __launch_bounds__(256)
gcn_scatter_kernel(const long long* __restrict__ rows,   // edges[0], int64
                   const long long* __restrict__ cols,   // edges[1], int64
                   const float*     __restrict__ ew,     // [E]
                   const float*     __restrict__ H,      // [n_nodes, 64]
                   float*           __restrict__ out,    // [n_nodes, 64]
                   int n_edges) {
    const int wavesPerBlock = 256 >> 5;                  // 8
    int edge = blockIdx.x * wavesPerBlock + (threadIdx.x >> 5);
    if (edge >= n_edges) return;
    const int lane = threadIdx.x & 31;

    const long long r    = rows[edge];
    const long long cidx = cols[edge];
    const float     w    = ew[edge];

    const float* hsrc = H   + (size_t)cidx * OUT_DIM;
    float*       dst  = out + (size_t)r    * OUT_DIM;

    float2 hv = *reinterpret_cast<const float2*>(hsrc + lane * 2);
    atomicAdd(dst + lane * 2 + 0, w * hv.x);             // global_atomic_add_f32
    atomicAdd(dst + lane * 2 + 1, w * hv.y);
}

// ---------------------------------------------------------------------------
extern "C" void kernel_launch(void* const* d_in, const int* in_sizes, int n_in,
                              void* d_out, int out_size, void* d_ws, size_t ws_size,
                              hipStream_t stream) {
    const float*     node_emb = (const float*)d_in[0];      // [N, 64] fp32
    const long long* edges    = (const long long*)d_in[1];  // [2, E]  int64
    const float*     ew       = (const float*)d_in[2];      // [E]     fp32
    const float*     W        = (const float*)d_in[3];      // [64,64] fp32

    const int n_nodes = in_sizes[0] / IN_DIM;
    const int n_edges = in_sizes[2];

    const long long* rows = edges;                // edges[0, :]
    const long long* cols = edges + n_edges;      // edges[1, :]

    float* out = (float*)d_out;
    float* H   = (float*)d_ws;                    // [n_nodes, 64] scratch (12.8 MB)

    // 1) out = 0  (float4 stores)
    {
        int n4 = out_size / 4;
        int blocks = (n4 + 255) / 256;
        zero_out_kernel<<<blocks, 256, 0, stream>>>((float4*)out, n4);
    }
    // 2) H = X @ W^T  (WMMA; each wave does 16 rows x all 64 cols)
    {
        int rowsPerBlock = 4 * 16;                // 4 waves x 16 rows
        int blocks = (n_nodes + rowsPerBlock - 1) / rowsPerBlock;  // 782
        gcn_gemm_wmma_kernel<<<blocks, 128, 0, stream>>>(node_emb, W, H, n_nodes);
    }
    // 3) scatter-add over edges (wave per edge)
    {
        int wavesPerBlock = 8;
        int blocks = (n_edges + wavesPerBlock - 1) / wavesPerBlock;
        gcn_scatter_kernel<<<blocks, 256, 0, stream>>>(rows, cols, ew, H, out, n_edges);
    }
}